// FP8ToFP32Converter_23407571764186
// MI455X (gfx1250) — compile-verified
//
#include <hip/hip_runtime.h>
#include <stdint.h>

// Output floats are exact 0.0f / 1.0f bit indicators -> all logic is integer.
typedef float v4f __attribute__((ext_vector_type(4)));

// Build the FP32 bit pattern from the 8 FP8(E4M3) bit-floats.
// a = {s, e3, e2, e1}, b = {e0, m2, m1, m0}
__device__ __forceinline__ uint32_t pack_fp32_word(v4f a, v4f b) {
    uint32_t s  = a.x > 0.5f;
    uint32_t e3 = a.y > 0.5f;
    uint32_t e2 = a.z > 0.5f;
    uint32_t e1 = a.w > 0.5f;
    uint32_t e0 = b.x > 0.5f;
    uint32_t m2 = b.y > 0.5f;
    uint32_t m1 = b.z > 0.5f;
    uint32_t m0 = b.w > 0.5f;

    uint32_t e = (e3 << 3) | (e2 << 2) | (e1 << 1) | e0;
    uint32_t m = (m2 << 2) | (m1 << 1) | m0;

    // normal:      exp = e + 120, mant = m2 m1 m0 0...
    // subnorm 1xx: exp = 120,     mant = m1 m0 0...
    // subnorm 01x: exp = 119,     mant = m0 0...
    // subnorm 001: exp = 118,     mant = 0
    // zero:        exp = 0,       mant = 0
    uint32_t exp8 = e  ? (e + 120u)
                  : (m2 ? 120u
                  : (m1 ? 119u
                  : (m0 ? 118u : 0u)));
    uint32_t mant = e  ? (m << 20)
                  : (m2 ? ((m & 3u) << 21)
                  : (m1 ? ((m & 1u) << 22) : 0u));

    return (s << 31) | (exp8 << 23) | mant;
}

// Wave-transposed store of one 32-row output block (4 KiB): every
// global_store_b128 instruction writes a contiguous 512 B span.
// Output float index within block: idx = k*128 + lane*4 + c
//   source row = k*4 + (lane>>3);  source bit = 31 - ((lane&7)*4 + c)
__device__ __forceinline__ void store_wave_block(uint32_t word, float* wbase,
                                                 unsigned lane) {
    const unsigned j0 = (lane & 7u) * 4u;
    #pragma unroll
    for (int k = 0; k < 8; ++k) {
        uint32_t w = (uint32_t)__shfl((int)word, (int)(k * 4 + (lane >> 3)), 32);
        v4f o;
        o.x = (float)((w >> (31u - (j0 + 0u))) & 1u);
        o.y = (float)((w >> (31u - (j0 + 1u))) & 1u);
        o.z = (float)((w >> (31u - (j0 + 2u))) & 1u);
        o.w = (float)((w >> (31u - (j0 + 3u))) & 1u);
        // 512 MiB streamed once -> non-temporal, don't rinse L2.
        __builtin_nontemporal_store(o, (v4f*)(wbase + (size_t)k * 128u + lane * 4u));
    }
}

__global__ __launch_bounds__(256)
void fp8bits_to_fp32bits_kernel(const float* __restrict__ in,
                                float* __restrict__ out,
                                unsigned nrows)
{
    // 16 KiB: 8 waves * 2 KiB (64 rows * 32 B per wave).
    __shared__ __align__(16) float tile[4096];

    const unsigned tid  = threadIdx.x;
    const unsigned lane = tid & 31u;
    const unsigned wv   = tid >> 5;
    const unsigned base = blockIdx.x * 512u + wv * 64u;  // first of 64 rows/wave

    if (base + 64u <= nrows) {
        // ---- CDNA5 async global->LDS staging: four coalesced 512 B sweeps,
        // 4 outstanding async b128 loads before a single ASYNCcnt wait. ----
        uint32_t ldsOff = (uint32_t)(uintptr_t)(&tile[0]) + wv * 2048u + lane * 16u;
        const char* gAddr = (const char*)(in + (size_t)base * 8u) + lane * 16u;
        asm volatile(
            "global_load_async_to_lds_b128 %0, %1, off\n\t"
            "global_load_async_to_lds_b128 %0, %1, off offset:512\n\t"
            "global_load_async_to_lds_b128 %0, %1, off offset:1024\n\t"
            "global_load_async_to_lds_b128 %0, %1, off offset:1536\n\t"
            "s_wait_asynccnt 0"
            :: "v"(ldsOff), "v"(gAddr)
            : "memory");
        // Only this wave's async ops wrote this wave's slab -> asynccnt wait
        // suffices, no workgroup barrier needed.

        // Each lane reads its two 32 B rows back (ds_load_b128 x4).
        const char* lbase = (const char*)(&tile[0]) + wv * 2048u + lane * 32u;
        const v4f* r0 = (const v4f*)lbase;             // row base+lane
        const v4f* r1 = (const v4f*)(lbase + 1024u);   // row base+32+lane
        v4f a0 = r0[0], b0 = r0[1];
        v4f a1 = r1[0], b1 = r1[1];

        uint32_t w0 = pack_fp32_word(a0, b0);
        uint32_t w1 = pack_fp32_word(a1, b1);

        store_wave_block(w0, out + (size_t)base * 32u, lane);
        store_wave_block(w1, out + (size_t)(base + 32u) * 32u, lane);
    } else {
        // Tail (dead for 4,194,304 rows = 8192 full blocks; kept for safety).
        #pragma unroll
        for (unsigned slab = 0; slab < 2; ++slab) {
            unsigned r = base + slab * 32u + lane;
            if (r < nrows) {
                const v4f* src = (const v4f*)(in + (size_t)r * 8u);
                v4f a = src[0];
                v4f b = src[1];
                uint32_t word = pack_fp32_word(a, b);
                float* dst = out + (size_t)r * 32u;
                #pragma unroll
                for (int j = 0; j < 32; ++j) {
                    dst[j] = (float)((word >> (31 - j)) & 1u);
                }
            }
        }
    }
}

extern "C" void kernel_launch(void* const* d_in, const int* in_sizes, int n_in,
                              void* d_out, int out_size, void* d_ws, size_t ws_size,
                              hipStream_t stream) {
    (void)n_in; (void)out_size; (void)d_ws; (void)ws_size;
    const float* in  = (const float*)d_in[0];
    float*       out = (float*)d_out;
    unsigned rows = (unsigned)(in_sizes[0] / 8);     // 4,194,304
    unsigned grid = (rows + 511u) / 512u;            // 8,192 blocks, 64 rows/wave
    fp8bits_to_fp32bits_kernel<<<grid, 256, 0, stream>>>(in, out, rows);
}